// DACRVQBottleneck_79577154060465
// MI455X (gfx1250) — compile-verified
//
#include <hip/hip_runtime.h>

// Problem constants
#define BB 8
#define DD 1024
#define TT 2048
#define NQ 9
#define CS 1024
#define CD 8
#define NT (BB*TT)            // 16384 rows (b*T + t)
#define Z_N (BB*DD*TT)        // 16777216
#define CODES_N (BB*NQ*TT)    // 147456
#define WSTRIDE 1028          // padded LDS stride for [8][1024] weights
#define TSTRIDE 17            // padded LDS stride for 16x16 staged tiles

typedef float v2f __attribute__((ext_vector_type(2)));
typedef float v8f __attribute__((ext_vector_type(8)));

// D(16x16 f32) = A(16x4 f32) * B(4x16 f32) + C
__device__ __forceinline__ v8f wmma4(v2f a, v2f b, v8f c) {
  return __builtin_amdgcn_wmma_f32_16x16x4_f32(false, a, false, b, (short)0, c,
                                               false, false);
}

// ---------------- init: residual = x ----------------
__global__ void k_copy(const float4* __restrict__ src, float4* __restrict__ dst,
                       int n4) {
  int i = blockIdx.x * blockDim.x + threadIdx.x;
  if (i < n4) dst[i] = src[i];
}

// ---------------- normalize all codebooks ----------------
__global__ void k_normcb(const float* __restrict__ cb, float* __restrict__ cbn) {
  int r = blockIdx.x * blockDim.x + threadIdx.x;  // row over NQ*CS
  if (r >= NQ * CS) return;
  const float4* s = (const float4*)(cb + (size_t)r * CD);
  float4 a = s[0], b = s[1];
  float n2 = a.x*a.x + a.y*a.y + a.z*a.z + a.w*a.w
           + b.x*b.x + b.y*b.y + b.z*b.z + b.w*b.w;
  float m = fmaxf(sqrtf(n2), 1e-12f);
  float4 oa = {a.x/m, a.y/m, a.z/m, a.w/m};
  float4 ob = {b.x/m, b.y/m, b.z/m, b.w/m};
  float4* d = (float4*)(cbn + (size_t)r * CD);
  d[0] = oa; d[1] = ob;
}

// ---------------- standalone in_proj (iteration 0 only) ----------------
// Ze[b*T+t][c] = sum_d Wi[c][d]*R[b][d][t] + bi[c].  M=8(dup16) K=1024 N=T.
__global__ __launch_bounds__(128) void k_inproj(const float* __restrict__ R,
    const float* __restrict__ Wi, const float* __restrict__ bi,
    float* __restrict__ Ze) {
  __shared__ float sW[CD * WSTRIDE];
  const int tid  = threadIdx.x;
  const int lane = tid & 31, wave = tid >> 5;
  const int tb = blockIdx.x & 31;              // T/64 = 32 tiles per batch
  const int b  = blockIdx.x >> 5;
  for (int idx = tid; idx < CD * DD; idx += 128) {
    int c = idx >> 10, k = idx & 1023;
    sW[c * WSTRIDE + k] = Wi[idx];
  }
  __syncthreads();
  const int half = lane >> 4;
  const int lid  = lane & 15;
  const int t0 = tb * 64 + wave * 16;
  const float* Rb = R + (size_t)b * DD * TT + t0 + lid;
  const float* wrow = &sW[(lid & 7) * WSTRIDE];  // rows 8-15 duplicate 0-7:
                                                 // garbage D rows are discarded
  v8f acc = {};
  int kk = half * 2;
  v2f a, bv, aN, bN;
  a.x = wrow[kk]; a.y = wrow[kk + 1];
  bv.x = Rb[(size_t)kk * TT]; bv.y = Rb[(size_t)(kk + 1) * TT];
#pragma unroll 4
  for (int k0 = 0; k0 < DD - 4; k0 += 4) {     // software pipelined
    int kn = k0 + 4 + half * 2;
    aN.x = wrow[kn]; aN.y = wrow[kn + 1];
    bN.x = Rb[(size_t)kn * TT]; bN.y = Rb[(size_t)(kn + 1) * TT];
    acc = wmma4(a, bv, acc);
    a = aN; bv = bN;
  }
  acc = wmma4(a, bv, acc);
  if (half == 0) {                             // C: lanes 0-15, VGPR v=(M=v,N=lid)
    size_t row = (size_t)b * TT + t0 + lid;
    float4 o0, o1;
    o0.x = acc[0] + bi[0]; o0.y = acc[1] + bi[1];
    o0.z = acc[2] + bi[2]; o0.w = acc[3] + bi[3];
    o1.x = acc[4] + bi[4]; o1.y = acc[5] + bi[5];
    o1.z = acc[6] + bi[6]; o1.w = acc[7] + bi[7];
    float4* dst = (float4*)(Ze + row * CD);
    dst[0] = o0; dst[1] = o1;
  }
}

// ---------------- codebook search + gather + loss partials ----------------
__global__ __launch_bounds__(256) void k_search(
    const float* __restrict__ Ze, const float* __restrict__ cbn_i,
    const float* __restrict__ cb_i, float* __restrict__ Zq,
    float* __restrict__ codes_out, float* __restrict__ partials, int iter) {
  __shared__ float sC[CS * 9];
  __shared__ float sred[256];
  const int tid = threadIdx.x;
  for (int idx = tid; idx < CS * CD; idx += 256) {
    int n = idx >> 3, k = idx & 7;
    sC[n * 9 + k] = cbn_i[idx];
  }
  __syncthreads();
  const int lane = tid & 31, wave = tid >> 5;
  const int half = lane >> 4, lid = lane & 15;
  const int rowBase = blockIdx.x * 128 + wave * 16;
  const float4* zr = (const float4*)(Ze + (size_t)(rowBase + lid) * CD);
  float4 z0 = zr[0], z1 = zr[1];
  v2f a0, a1;
  a0.x = half ? z0.z : z0.x;  a0.y = half ? z0.w : z0.y;
  a1.x = half ? z1.z : z1.x;  a1.y = half ? z1.w : z1.y;
  float best[8]; int bidx[8];
#pragma unroll
  for (int v = 0; v < 8; v++) { best[v] = -3.402823466e38f; bidx[v] = 0; }
  for (int n0 = 0; n0 < CS; n0 += 16) {
    const float* crow = &sC[(n0 + lid) * 9];
    v2f b0, b1;
    b0.x = crow[half * 2];     b0.y = crow[half * 2 + 1];
    b1.x = crow[4 + half * 2]; b1.y = crow[4 + half * 2 + 1];
    v8f acc = {};
    acc = wmma4(a0, b0, acc);
    acc = wmma4(a1, b1, acc);
    int code = n0 + lid;
#pragma unroll
    for (int v = 0; v < 8; v++) {
      float s = acc[v];
      if (s > best[v]) { best[v] = s; bidx[v] = code; }  // strict > keeps first
    }
  }
#pragma unroll
  for (int m = 1; m <= 8; m <<= 1) {           // reduce 16 cols, stays in half
#pragma unroll
    for (int v = 0; v < 8; v++) {
      float os = __shfl_xor(best[v], m, 32);
      int   oi = __shfl_xor(bidx[v], m, 32);
      if (os > best[v] || (os == best[v] && oi < bidx[v])) {
        best[v] = os; bidx[v] = oi;
      }
    }
  }
  float partial = 0.f;
  if (lid < 8) {
    int myIdx = 0;
#pragma unroll
    for (int v = 0; v < 8; v++) if (lid == v) myIdx = bidx[v];
    int r = rowBase + half * 8 + lid;
    int bb = r >> 11;
    int t  = r & (TT - 1);
    codes_out[(size_t)bb * NQ * TT + (size_t)iter * TT + t] = (float)myIdx;
    const float4* cr = (const float4*)(cb_i + (size_t)myIdx * CD);
    float4 c0 = cr[0], c1 = cr[1];
    float4* qd = (float4*)(Zq + (size_t)r * CD);
    qd[0] = c0; qd[1] = c1;
    const float4* er = (const float4*)(Ze + (size_t)r * CD);
    float4 e0 = er[0], e1 = er[1];
    float d0 = e0.x - c0.x, d1 = e0.y - c0.y, d2 = e0.z - c0.z, d3 = e0.w - c0.w;
    float d4 = e1.x - c1.x, d5 = e1.y - c1.y, d6 = e1.z - c1.z, d7 = e1.w - c1.w;
    partial = d0*d0 + d1*d1 + d2*d2 + d3*d3 + d4*d4 + d5*d5 + d6*d6 + d7*d7;
  }
  sred[tid] = partial;
  __syncthreads();
  for (int s = 128; s > 0; s >>= 1) {
    if (tid < s) sred[tid] += sred[tid + s];
    __syncthreads();
  }
  if (tid == 0) partials[iter * 128 + blockIdx.x] = sred[0];
}

// ---- fused: R -= Wout(i)*zq + bo(i)   AND   Ze = Win(i+1)*Rnew + bin(i+1) ----
// Per wave: one 16-wide t strip. d-loop computes the out_proj 16x16 tile,
// RMWs R, stages Rnew in LDS (C-layout -> B-layout relay), then 4 more WMMAs
// accumulate the next iteration's z_e. 6 WMMAs per d-tile.
__global__ __launch_bounds__(256) void k_fused(
    float* __restrict__ R, const float* __restrict__ Wo,
    const float* __restrict__ bo, const float* __restrict__ Zq,
    const float* __restrict__ Wi1, const float* __restrict__ bi1,
    float* __restrict__ Ze) {
  __shared__ float sWi[CD * WSTRIDE];          // next-iter in_proj weights
  __shared__ float sBo[DD];
  __shared__ float sT[8 * 16 * TSTRIDE];       // per-wave 16x16 Rnew tiles
  const int tid = threadIdx.x;
  for (int idx = tid; idx < CD * DD; idx += 256) {
    int c = idx >> 10, k = idx & 1023;
    sWi[c * WSTRIDE + k] = Wi1[idx];
  }
  for (int idx = tid; idx < DD; idx += 256) sBo[idx] = bo[idx];
  __syncthreads();
  const int lane = tid & 31, wave = tid >> 5;
  const int half = lane >> 4, lid = lane & 15;
  const int b  = blockIdx.x >> 4;
  const int t0 = (blockIdx.x & 15) * 128 + wave * 16;
  float* Rb = R + (size_t)b * DD * TT + t0 + lid;   // column base for this lane
  float* tile = &sT[wave * 16 * TSTRIDE];
  const float* wirow = &sWi[(lid & 7) * WSTRIDE];
  // B operand of out_proj (z_q row), d-invariant
  const float4* qr = (const float4*)(Zq + ((size_t)b * TT + t0 + lid) * CD);
  float4 q0 = qr[0], q1 = qr[1];
  v2f b0o, b1o;
  b0o.x = half ? q0.z : q0.x;  b0o.y = half ? q0.w : q0.y;
  b1o.x = half ? q1.z : q1.x;  b1o.y = half ? q1.w : q1.y;
  const float4* wo4 = (const float4*)Wo;
  v8f zacc = {};
  for (int d0 = 0; d0 < DD; d0 += 16) {
    // --- out_proj tile: A = Wout rows d0+lid (2x b128, select by half) ---
    float4 w0 = wo4[(size_t)(d0 + lid) * 2];
    float4 w1 = wo4[(size_t)(d0 + lid) * 2 + 1];
    v2f a0, a1;
    a0.x = half ? w0.z : w0.x;  a0.y = half ? w0.w : w0.y;
    a1.x = half ? w1.z : w1.x;  a1.y = half ? w1.w : w1.y;
    v8f acc = {};
    acc = wmma4(a0, b0o, acc);
    acc = wmma4(a1, b1o, acc);
    // --- RMW residual, stage Rnew tile in LDS ---
#pragma unroll
    for (int v = 0; v < 8; v++) {
      int d = d0 + half * 8 + v;               // C-layout row for this half
      size_t off = (size_t)d * TT;
      float rn = Rb[off] - acc[v] - sBo[d];
      Rb[off] = rn;
      tile[(half * 8 + v) * TSTRIDE + lid] = rn;
    }
    // --- z_e(i+1) accumulation over this d-tile: K=16 -> 4 WMMAs ---
#pragma unroll
    for (int k0 = 0; k0 < 16; k0 += 4) {
      int kk = k0 + half * 2;
      v2f a, bv;
      a.x  = wirow[d0 + kk];  a.y = wirow[d0 + kk + 1];
      bv.x = tile[kk * TSTRIDE + lid];
      bv.y = tile[(kk + 1) * TSTRIDE + lid];
      zacc = wmma4(a, bv, zacc);
    }
  }
  if (half == 0) {
    size_t row = (size_t)b * TT + t0 + lid;
    float4 o0, o1;
    o0.x = zacc[0] + bi1[0]; o0.y = zacc[1] + bi1[1];
    o0.z = zacc[2] + bi1[2]; o0.w = zacc[3] + bi1[3];
    o1.x = zacc[4] + bi1[4]; o1.y = zacc[5] + bi1[5];
    o1.z = zacc[6] + bi1[6]; o1.w = zacc[7] + bi1[7];
    float4* dst = (float4*)(Ze + row * CD);
    dst[0] = o0; dst[1] = o1;
  }
}

// ---------------- last-iteration out_proj (no z_e needed) ----------------
__global__ __launch_bounds__(256) void k_outproj(
    float* __restrict__ R, const float* __restrict__ Wo,
    const float* __restrict__ bo, const float* __restrict__ Zq) {
  __shared__ float sBo[DD];
  const int tid = threadIdx.x;
  for (int idx = tid; idx < DD; idx += 256) sBo[idx] = bo[idx];
  __syncthreads();
  const int lane = tid & 31, wave = tid >> 5;
  const int half = lane >> 4, lid = lane & 15;
  const int b  = blockIdx.x >> 4;
  const int t0 = (blockIdx.x & 15) * 128 + wave * 16;
  float* Rb = R + (size_t)b * DD * TT + t0 + lid;
  const float4* qr = (const float4*)(Zq + ((size_t)b * TT + t0 + lid) * CD);
  float4 q0 = qr[0], q1 = qr[1];
  v2f b0o, b1o;
  b0o.x = half ? q0.z : q0.x;  b0o.y = half ? q0.w : q0.y;
  b1o.x = half ? q1.z : q1.x;  b1o.y = half ? q1.w : q1.y;
  const float4* wo4 = (const float4*)Wo;
  for (int d0 = 0; d0 < DD; d0 += 16) {
    float4 w0 = wo4[(size_t)(d0 + lid) * 2];
    float4 w1 = wo4[(size_t)(d0 + lid) * 2 + 1];
    v2f a0, a1;
    a0.x = half ? w0.z : w0.x;  a0.y = half ? w0.w : w0.y;
    a1.x = half ? w1.z : w1.x;  a1.y = half ? w1.w : w1.y;
    v8f acc = {};
    acc = wmma4(a0, b0o, acc);
    acc = wmma4(a1, b1o, acc);
#pragma unroll
    for (int v = 0; v < 8; v++) {
      int d = d0 + half * 8 + v;
      size_t off = (size_t)d * TT;
      Rb[off] = Rb[off] - acc[v] - sBo[d];
    }
  }
}

// ---------------- finalize: z = x - residual; losses ----------------
__global__ void k_final(const float4* __restrict__ x4, float4* __restrict__ z4,
                        const float* __restrict__ partials,
                        float* __restrict__ loss_out) {
  int i = blockIdx.x * blockDim.x + threadIdx.x;
  if (i < Z_N / 4) {
    float4 xv = x4[i], rv = z4[i];
    float4 o = {xv.x - rv.x, xv.y - rv.y, xv.z - rv.z, xv.w - rv.w};
    z4[i] = o;
  }
  if (blockIdx.x == 0 && threadIdx.x == 0) {
    float s = 0.f;
    for (int j = 0; j < NQ * 128; j++) s += partials[j];
    float v = s / (float)((size_t)NQ * BB * CD * TT);
    loss_out[0] = v;   // commit_loss
    loss_out[1] = v;   // cb_loss (identical in forward)
  }
}

extern "C" void kernel_launch(void* const* d_in, const int* in_sizes, int n_in,
                              void* d_out, int out_size, void* d_ws,
                              size_t ws_size, hipStream_t stream) {
  const float* x    = (const float*)d_in[0];
  const float* wIn  = (const float*)d_in[1];
  const float* bIn  = (const float*)d_in[2];
  const float* wOut = (const float*)d_in[3];
  const float* bOut = (const float*)d_in[4];
  const float* cb   = (const float*)d_in[5];
  float* out   = (float*)d_out;
  float* Rres  = out;                    // z region doubles as residual buffer
  float* codes = out + Z_N;
  float* loss  = out + Z_N + CODES_N;
  float* ws  = (float*)d_ws;
  float* cbn = ws;                                   // NQ*CS*CD = 73728
  float* Ze  = ws + 73728;                           // NT*CD   = 131072
  float* Zq  = ws + 73728 + 131072;                  // NT*CD   = 131072
  float* partials = ws + 73728 + 2 * 131072;         // NQ*128

  k_copy<<<Z_N / 4 / 256, 256, 0, stream>>>((const float4*)x, (float4*)Rres,
                                            Z_N / 4);
  k_normcb<<<(NQ * CS + 255) / 256, 256, 0, stream>>>(cb, cbn);
  k_inproj<<<BB * (TT / 64), 128, 0, stream>>>(Rres, wIn, bIn, Ze);
  for (int i = 0; i < NQ; i++) {
    k_search<<<NT / 128, 256, 0, stream>>>(
        Ze, cbn + (size_t)i * CS * CD, cb + (size_t)i * CS * CD, Zq, codes,
        partials, i);
    if (i < NQ - 1) {
      k_fused<<<BB * 16, 256, 0, stream>>>(
          Rres, wOut + (size_t)i * DD * CD, bOut + (size_t)i * DD, Zq,
          wIn + (size_t)(i + 1) * CD * DD, bIn + (size_t)(i + 1) * CD, Ze);
    } else {
      k_outproj<<<BB * 16, 256, 0, stream>>>(
          Rres, wOut + (size_t)i * DD * CD, bOut + (size_t)i * DD, Zq);
    }
  }
  k_final<<<Z_N / 4 / 256, 256, 0, stream>>>((const float4*)x, (float4*)Rres,
                                             partials, loss);
}